// Gemma4VisionPooler_57775900066507
// MI455X (gfx1250) — compile-verified
//
#include <hip/hip_runtime.h>
#include <hip/hip_bf16.h>
#include <math.h>

// Problem constants (from the reference).
#define B_ 8
#define S_ 4096
#define H_ 1152
#define L_ 256
#define NTHREADS 288          // 9 wave32 waves; 288*4 floats == 1152 == H
#define CAP 128               // per-segment row-list capacity (mean 16, 12+ sigma safe)
#define CHUNK 15              // ceil(4096/288); contiguous per-thread scan range
#define WCAP 512              // per-16-segment-group row-list capacity (mean 256, 16 sigma)

#define INV16  0.0625f
#define ROOT_H 33.941125496954285f   // sqrt(1152) rounded to f32

// CDNA5 async global->LDS path, guarded so the file always compiles.
#if defined(__HIP_DEVICE_COMPILE__) && \
    __has_builtin(__builtin_amdgcn_global_load_async_to_lds_b128) && \
    __has_builtin(__builtin_amdgcn_s_wait_asynccnt)
#define USE_ASYNC_LDS 1
#else
#define USE_ASYNC_LDS 0
#endif

#if USE_ASYNC_LDS
typedef int v4i_ __attribute__((vector_size(16)));   // int32x4, matches builtin param
__device__ __forceinline__ void async_copy_b128(const float* gsrc, float4* ldst) {
  // One instruction per wave: each lane moves 16 bytes global -> LDS.
  __builtin_amdgcn_global_load_async_to_lds_b128((v4i_*)gsrc, (v4i_*)ldst, 0, 0);
}
#endif

// ============================================================================
// Kernel 1: segment gather-sum with async global->LDS double-buffered pipeline.
// One block per (batch, segment). Memory-roofline optimal (~160 MB ~ 7 us).
// ============================================================================
__global__ __launch_bounds__(NTHREADS)
void Gemma4VisionPooler_pool_kernel(const float* __restrict__ hidden,
                                    const int* __restrict__ pos,   // (B,S,2) int32
                                    float* __restrict__ out) {
  const int tid = threadIdx.x;
  const int blk = blockIdx.x;     // blk = b*256 + seg
  const int b   = blk >> 8;
  const int seg = blk & 255;

  __shared__ int    sh[NTHREADS];     // reused: max-reduce, then counts->offsets
  __shared__ int    s_list[CAP];      // ascending row indices of this segment
  __shared__ int    s_meta[2];        // [0] = q = max_x//4, [1] = total count
  __shared__ float4 lbuf[2][NTHREADS];// double-buffered row staging (2 x 4608 B)

  const int* pb = pos + (size_t)b * (S_ * 2);

  // ---- scan 1: per-batch max of clamped x --------------------------------
  int mx = 0;
  for (int s = tid; s < S_; s += NTHREADS) {
    int x = pb[2 * s];
    mx = x > mx ? x : mx;             // clamp(min=0) folded in (mx starts at 0)
  }
  sh[tid] = mx;
  __syncthreads();
  if (tid == 0) {
    int m = 0;
    for (int i = 0; i < NTHREADS; ++i) m = sh[i] > m ? sh[i] : m;
    s_meta[0] = (m + 1) >> 2;         // q = (max_x) // 4 with max_x = max+1
  }
  __syncthreads();
  const int q = s_meta[0];

  // ---- scan 2: per-thread match counts over contiguous chunks ------------
  const int s0 = tid * CHUNK;
  const int s1 = (s0 + CHUNK < S_) ? (s0 + CHUNK) : S_;
  int cnt = 0;
  for (int s = s0; s < s1; ++s) {
    int x = pb[2 * s];     x = x > 0 ? x : 0;
    int y = pb[2 * s + 1]; y = y > 0 ? y : 0;
    int idx = (x >> 2) + q * (y >> 2);
    cnt += (idx == seg);
  }
  sh[tid] = cnt;
  __syncthreads();
  if (tid == 0) {                      // exclusive prefix scan (288 entries)
    int run = 0;
    for (int i = 0; i < NTHREADS; ++i) { int c = sh[i]; sh[i] = run; run += c; }
    s_meta[1] = run;
  }
  __syncthreads();
  int off = sh[tid];
  const int n_total = s_meta[1];
  const int n = n_total < CAP ? n_total : CAP;

  // ---- scan 3: write matching row ids in ascending s order ---------------
  for (int s = s0; s < s1; ++s) {
    int x = pb[2 * s];     x = x > 0 ? x : 0;
    int y = pb[2 * s + 1]; y = y > 0 ? y : 0;
    int idx = (x >> 2) + q * (y >> 2);
    if (idx == seg) {
      if (off < CAP) s_list[off] = s;
      ++off;
    }
  }
  __syncthreads();

  // ---- gather + sum the segment's rows (memory-bound hot loop) -----------
  float4 acc = make_float4(0.f, 0.f, 0.f, 0.f);

#if USE_ASYNC_LDS
  if (n > 0) {
    {
      const float* src = hidden + ((size_t)b * S_ + s_list[0]) * H_ + tid * 4;
      async_copy_b128(src, &lbuf[0][tid]);
    }
    for (int r = 0; r < n; ++r) {
      if (r + 1 < n) {
        const float* src = hidden + ((size_t)b * S_ + s_list[r + 1]) * H_ + tid * 4;
        async_copy_b128(src, &lbuf[(r + 1) & 1][tid]);
        __builtin_amdgcn_s_wait_asynccnt(1);   // oldest copy (row r) complete
      } else {
        __builtin_amdgcn_s_wait_asynccnt(0);   // drain on the tail
      }
      asm volatile("" ::: "memory");           // keep the LDS read below the wait
      float4 v = lbuf[r & 1][tid];             // each lane reads only its own 16 B
      acc.x += v.x; acc.y += v.y; acc.z += v.z; acc.w += v.w;
    }
  }
#else
  for (int r = 0; r < n; ++r) {
    if (r + 1 < n) {
      __builtin_prefetch(hidden + ((size_t)b * S_ + s_list[r + 1]) * H_ + tid * 4, 0, 3);
    }
    const float4* src =
        (const float4*)(hidden + ((size_t)b * S_ + s_list[r]) * H_);
    float4 v = src[tid];
    acc.x += v.x; acc.y += v.y; acc.z += v.z; acc.w += v.w;
  }
#endif

  // ---- scale exactly like the reference: (sum/16) * sqrt(1152) -----------
  acc.x = (acc.x * INV16) * ROOT_H;
  acc.y = (acc.y * INV16) * ROOT_H;
  acc.z = (acc.z * INV16) * ROOT_H;
  acc.w = (acc.w * INV16) * ROOT_H;

  float4* outp = (float4*)(out + (size_t)blk * H_);
  outp[tid] = acc;

  if (tid == 0) {
    out[(size_t)B_ * L_ * H_ + blk] = (n_total > 0) ? 1.0f : 0.0f;
  }
}

// ============================================================================
// Kernel 2: the same pooling expressed as the one-hot GEMM on the WMMA path.
//   D(16 segs x 16 cols) += A(16x4 one-hot membership) x B(4 rows x 16 cols)
// using V_WMMA_F32_16X16X4_F32 (full fp32 matrix pipe). One block per
// (batch, group of 16 segments); each hidden row is still read exactly once
// globally, so this sits on the same 23.3 TB/s roofline as kernel 1.
// ============================================================================
typedef float v2f_ __attribute__((ext_vector_type(2)));
typedef float v8f_ __attribute__((ext_vector_type(8)));

__global__ __launch_bounds__(NTHREADS)
void Gemma4VisionPooler_pool_wmma_kernel(const float* __restrict__ hidden,
                                         const int* __restrict__ pos,
                                         float* __restrict__ out) {
  const int tid  = threadIdx.x;
  const int lane = tid & 31;
  const int wave = tid >> 5;        // 0..8
  const int blk  = blockIdx.x;      // 128 blocks: b*16 + g
  const int b    = blk >> 4;
  const int g    = blk & 15;        // segments [16g, 16g+16)

  __shared__ int sh[NTHREADS];
  __shared__ int s_rows[WCAP];      // union row list (ascending), padded to x4
  __shared__ int s_seg[WCAP];       // seg-in-group (0..15), 16 = padding row
  __shared__ int s_cnt[16];         // per-segment member counts (for the mask)
  __shared__ int s_meta[2];         // [0]=q, [1]=total count

  const int* pb = pos + (size_t)b * (S_ * 2);
  const int seg_lo = g << 4;

  if (tid < 16) s_cnt[tid] = 0;

  // ---- per-batch max of clamped x ----------------------------------------
  int mx = 0;
  for (int s = tid; s < S_; s += NTHREADS) {
    int x = pb[2 * s];
    mx = x > mx ? x : mx;
  }
  sh[tid] = mx;
  __syncthreads();
  if (tid == 0) {
    int m = 0;
    for (int i = 0; i < NTHREADS; ++i) m = sh[i] > m ? sh[i] : m;
    s_meta[0] = (m + 1) >> 2;
  }
  __syncthreads();
  const int q = s_meta[0];

  // ---- count matches for this 16-segment group (contiguous chunks) -------
  const int s0 = tid * CHUNK;
  const int s1 = (s0 + CHUNK < S_) ? (s0 + CHUNK) : S_;
  int cnt = 0;
  for (int s = s0; s < s1; ++s) {
    int x = pb[2 * s];     x = x > 0 ? x : 0;
    int y = pb[2 * s + 1]; y = y > 0 ? y : 0;
    int idx = (x >> 2) + q * (y >> 2);
    cnt += (unsigned)(idx - seg_lo) < 16u;
  }
  sh[tid] = cnt;
  __syncthreads();
  if (tid == 0) {
    int run = 0;
    for (int i = 0; i < NTHREADS; ++i) { int c = sh[i]; sh[i] = run; run += c; }
    s_meta[1] = run;
  }
  __syncthreads();
  int off = sh[tid];
  const int n_total = s_meta[1];
  const int n = n_total < WCAP ? n_total : WCAP;

  // ---- write row ids + segment-in-group, ascending ------------------------
  for (int s = s0; s < s1; ++s) {
    int x = pb[2 * s];     x = x > 0 ? x : 0;
    int y = pb[2 * s + 1]; y = y > 0 ? y : 0;
    int idx = (x >> 2) + q * (y >> 2);
    int d = idx - seg_lo;
    if ((unsigned)d < 16u) {
      if (off < WCAP) { s_rows[off] = s; s_seg[off] = d; }
      ++off;
      atomicAdd(&s_cnt[d], 1);
    }
  }
  __syncthreads();
  const int n4 = (n + 3) >> 2;
  if (tid == 0) {                    // pad to a multiple of 4 with inert rows
    for (int i = n; i < n4 * 4 && i < WCAP; ++i) { s_rows[i] = 0; s_seg[i] = 16; }
  }
  __syncthreads();

  // ---- one-hot GEMM over the row list: 9 waves x 8 column tiles ----------
  const float* hb = hidden + (size_t)b * S_ * H_;
  const int m_lane = lane & 15;                    // A: M = lane&15
  const int k_off  = (lane < 16) ? 0 : 2;          // A/B: K pair per half-wave
  const int m_base = (lane < 16) ? 0 : 8;          // D: M = i (+8 for hi lanes)

  for (int t = wave * 8; t < wave * 8 + 8; ++t) {  // 72 = 9 waves * 8 tiles
    const int col = t * 16 + m_lane;               // B/D: N = lane&15
    v8f_ c = {0.f, 0.f, 0.f, 0.f, 0.f, 0.f, 0.f, 0.f};
    for (int kc = 0; kc < n4; ++kc) {
      const int k0 = kc * 4 + k_off;
      const int r0 = s_rows[k0], r1 = s_rows[k0 + 1];
      const int g0 = s_seg[k0],  g1 = s_seg[k0 + 1];
      v2f_ a;  a.x = (g0 == m_lane) ? 1.f : 0.f;   // one-hot membership
               a.y = (g1 == m_lane) ? 1.f : 0.f;
      v2f_ bb; bb.x = hb[(size_t)r0 * H_ + col];   // two hidden-row elements
               bb.y = hb[(size_t)r1 * H_ + col];
      c = __builtin_amdgcn_wmma_f32_16x16x4_f32(false, a, false, bb,
                                                (short)0, c, false, false);
    }
    // D layout: VGPR i -> (M = i + m_base, N = lane&15)
    for (int i = 0; i < 8; ++i) {
      float v = (c[i] * INV16) * ROOT_H;
      out[((size_t)b * L_ + seg_lo + m_base + i) * H_ + col] = v;
    }
  }

  if (tid < 16) {                    // this block's 16 mask entries
    out[(size_t)B_ * L_ * H_ + (size_t)b * L_ + seg_lo + tid] =
        (s_cnt[tid] > 0) ? 1.0f : 0.0f;
  }
}

extern "C" void kernel_launch(void* const* d_in, const int* in_sizes, int n_in,
                              void* d_out, int out_size, void* d_ws, size_t ws_size,
                              hipStream_t stream) {
  (void)in_sizes; (void)n_in; (void)d_ws; (void)ws_size; (void)out_size;
  const float* hidden = (const float*)d_in[0];
  const int*   pos    = (const int*)d_in[1];   // patch_positions (B,S,2)
  float* out = (float*)d_out;

  // Launched path: async-pipeline gather kernel (reference-order accumulation).
  // The WMMA one-hot-GEMM kernel above is the equivalent matrix-op formulation
  // (same bytes moved, same roofline); swap the launch below to use it:
  //   Gemma4VisionPooler_pool_wmma_kernel<<<dim3(B_*16), dim3(NTHREADS), 0, stream>>>(...)
  dim3 grid(B_ * L_);
  dim3 block(NTHREADS);
  hipLaunchKernelGGL(Gemma4VisionPooler_pool_kernel, grid, block, 0, stream,
                     hidden, pos, out);
}